// SimplePredictor_21612275433796
// MI455X (gfx1250) — compile-verified
//
#include <hip/hip_runtime.h>
#include <math.h>

#define B_    32
#define L_    2048
#define D_    1024
#define D2_   2048
#define EPS_  1e-5f
#define ROWS_ 64          // rows of x per block in the fused pooling pass

typedef float v2f __attribute__((ext_vector_type(2)));
typedef float v4f __attribute__((ext_vector_type(4)));
typedef float v8f __attribute__((ext_vector_type(8)));

// ---------------- ntgt: per-batch sum of int64 mask ----------------
__global__ void ntgt_kernel(const long long* __restrict__ mask, int* __restrict__ ntgt) {
  __shared__ int sd[256];
  int b = blockIdx.x;
  int s = 0;
  for (int l = threadIdx.x; l < L_; l += 256)
    s += (int)mask[(size_t)b * L_ + l];
  sd[threadIdx.x] = s;
  __syncthreads();
  for (int st = 128; st > 0; st >>= 1) {
    if (threadIdx.x < (unsigned)st) sd[threadIdx.x] += sd[threadIdx.x + st];
    __syncthreads();
  }
  if (threadIdx.x == 0) ntgt[b] = sd[0];
}

// ---------------- fused single pass over x ----------------
// grid (B, L/ROWS_), block 256. Thread t owns columns [4t, 4t+3] (float4 => b128 loads).
// Produces: partial column sums (atomic f32 add into colsum) and per-row nonzero
// bits -> one popcount atomicAdd into num_ctx[b] per block. x is touched ONCE.
__global__ void pool_fused_kernel(const float* __restrict__ x,
                                  float* __restrict__ colsum,
                                  int* __restrict__ num_ctx) {
  __shared__ unsigned sm0[256];
  __shared__ unsigned sm1[256];
  int b   = blockIdx.x;
  int r0  = blockIdx.y * ROWS_;
  int tid = threadIdx.x;

  v4f acc = {0.0f, 0.0f, 0.0f, 0.0f};
  unsigned m0 = 0u, m1 = 0u;

  const char* base = (const char*)(x + ((size_t)b * L_ + r0) * D_) + (size_t)tid * 16;
  for (int r = 0; r < ROWS_; ++r) {
    const v4f* p = (const v4f*)(base + (size_t)r * D_ * sizeof(float));
    v4f v = __builtin_nontemporal_load(p);      // streamed once: NT load
    acc += v;
    unsigned nz = ((v.x != 0.0f) | (v.y != 0.0f) | (v.z != 0.0f) | (v.w != 0.0f)) ? 1u : 0u;
    if (r < 32) m0 |= nz << r;
    else        m1 |= nz << (r - 32);
  }

  sm0[tid] = m0; sm1[tid] = m1;
  __syncthreads();
  for (int st = 128; st > 0; st >>= 1) {
    if (tid < (unsigned)st) { sm0[tid] |= sm0[tid + st]; sm1[tid] |= sm1[tid + st]; }
    __syncthreads();
  }
  if (tid == 0) {
    int cnt = __popc(sm0[0]) + __popc(sm1[0]);
    if (cnt) atomicAdd(&num_ctx[b], cnt);
  }

  float* cs = colsum + b * D_ + tid * 4;
  atomicAdd(cs + 0, acc.x);
  atomicAdd(cs + 1, acc.y);
  atomicAdd(cs + 2, acc.z);
  atomicAdd(cs + 3, acc.w);
}

// ---------------- WMMA GEMM1: mean[32,1024] @ W1[1024,2048] + b1, erf-GELU ----------------
__global__ void gemm1_gelu_kernel(const float* __restrict__ colsum,
                                  const int* __restrict__ num_ctx,
                                  const float* __restrict__ W1,
                                  const float* __restrict__ b1,
                                  float* __restrict__ h) {
  int wave = (int)((blockIdx.x * blockDim.x + threadIdx.x) >> 5);
  int lane = threadIdx.x & 31;
  const int tiles_n = D2_ / 16;         // 128
  int tm = wave / tiles_n;              // 0..1
  int tn = wave % tiles_n;
  int half = lane >> 4;                 // K sub-block (ISA A/B lane split)
  int idx  = lane & 15;
  int m = tm * 16 + idx;                // A row (batch) for this lane
  int n = tn * 16 + idx;                // B column for this lane
  float rd = 1.0f / (float)max(num_ctx[m], 1);

  v8f acc = {};
  for (int k0 = 0; k0 < D_; k0 += 4) {
    int ka = k0 + half * 2;
    v2f a, bb;
    a.x  = colsum[m * D_ + ka]     * rd;   // A[m][ka]
    a.y  = colsum[m * D_ + ka + 1] * rd;   // A[m][ka+1]
    bb.x = W1[(size_t)ka * D2_ + n];       // B[ka][n]
    bb.y = W1[(size_t)(ka + 1) * D2_ + n]; // B[ka+1][n]
    acc = __builtin_amdgcn_wmma_f32_16x16x4_f32(
        false, a, false, bb, (short)0, acc, false, false);
  }
  for (int v = 0; v < 8; ++v) {
    int row = tm * 16 + v + 8 * half;      // C/D: VGPR v -> M = v + 8*half
    int col = tn * 16 + idx;
    float xv = acc[v] + b1[col];
    float g  = 0.5f * xv * (1.0f + erff(xv * 0.70710678118654752f)); // exact-erf GELU
    h[row * D2_ + col] = g;
  }
}

// ---------------- LayerNorm over last dim (2048), in place ----------------
__global__ void layernorm_kernel(float* __restrict__ h,
                                 const float* __restrict__ gamma,
                                 const float* __restrict__ beta) {
  __shared__ float sdata[256];
  int b = blockIdx.x;
  float* row = h + (size_t)b * D2_;

  float s = 0.0f;
  for (int i = threadIdx.x; i < D2_; i += 256) s += row[i];
  sdata[threadIdx.x] = s;
  __syncthreads();
  for (int st = 128; st > 0; st >>= 1) {
    if (threadIdx.x < (unsigned)st) sdata[threadIdx.x] += sdata[threadIdx.x + st];
    __syncthreads();
  }
  float mu = sdata[0] * (1.0f / D2_);
  __syncthreads();

  float v = 0.0f;
  for (int i = threadIdx.x; i < D2_; i += 256) {
    float d = row[i] - mu;
    v += d * d;
  }
  sdata[threadIdx.x] = v;
  __syncthreads();
  for (int st = 128; st > 0; st >>= 1) {
    if (threadIdx.x < (unsigned)st) sdata[threadIdx.x] += sdata[threadIdx.x + st];
    __syncthreads();
  }
  float rstd = rsqrtf(sdata[0] * (1.0f / D2_) + EPS_);
  __syncthreads();

  for (int i = threadIdx.x; i < D2_; i += 256)
    row[i] = (row[i] - mu) * rstd * gamma[i] + beta[i];
}

// ---------------- WMMA GEMM2: hn[32,2048] @ W2[2048,1024] + b2 ----------------
__global__ void gemm2_kernel(const float* __restrict__ hn,
                             const float* __restrict__ W2,
                             const float* __restrict__ b2,
                             float* __restrict__ pred) {
  int wave = (int)((blockIdx.x * blockDim.x + threadIdx.x) >> 5);
  int lane = threadIdx.x & 31;
  const int tiles_n = D_ / 16;          // 64
  int tm = wave / tiles_n;
  int tn = wave % tiles_n;
  int half = lane >> 4;
  int idx  = lane & 15;
  int m = tm * 16 + idx;
  int n = tn * 16 + idx;

  v8f acc = {};
  for (int k0 = 0; k0 < D2_; k0 += 4) {
    int ka = k0 + half * 2;
    v2f a, bb;
    a.x  = hn[m * D2_ + ka];
    a.y  = hn[m * D2_ + ka + 1];
    bb.x = W2[(size_t)ka * D_ + n];
    bb.y = W2[(size_t)(ka + 1) * D_ + n];
    acc = __builtin_amdgcn_wmma_f32_16x16x4_f32(
        false, a, false, bb, (short)0, acc, false, false);
  }
  for (int v = 0; v < 8; ++v) {
    int row = tm * 16 + v + 8 * half;
    int col = tn * 16 + idx;
    pred[row * D_ + col] = acc[v] + b2[col];
  }
}

// ---------------- scatter: out[b,t,:] = (t < ntgt[b]) ? pred[b,:] : 0 (NT b128 stores) ----
__global__ void scatter_kernel(const float* __restrict__ pred,
                               const int* __restrict__ ntgt,
                               v4f* __restrict__ out, int T, int total4) {
  int i = blockIdx.x * 256 + threadIdx.x;
  if (i >= total4) return;
  int e  = i << 2;                       // element index
  int d  = e & (D_ - 1);
  int bt = e >> 10;                      // / D_
  int t  = bt % T;
  int b  = bt / T;
  v4f v = {0.0f, 0.0f, 0.0f, 0.0f};
  if (t < ntgt[b]) v = *(const v4f*)(pred + b * D_ + d);
  __builtin_nontemporal_store(v, &out[i]);
}

// ---------------- launch ----------------
extern "C" void kernel_launch(void* const* d_in, const int* in_sizes, int n_in,
                              void* d_out, int out_size, void* d_ws, size_t ws_size,
                              hipStream_t stream) {
  const float*     x     = (const float*)d_in[0];
  const long long* mask  = (const long long*)d_in[1];
  const float*     W1    = (const float*)d_in[2];
  const float*     b1    = (const float*)d_in[3];
  const float*     gamma = (const float*)d_in[4];
  const float*     beta  = (const float*)d_in[5];
  const float*     W2    = (const float*)d_in[6];
  const float*     b2    = (const float*)d_in[7];

  char* ws = (char*)d_ws;
  float* colsum = (float*)ws;  ws += (size_t)B_ * D_  * sizeof(float);
  int*   numctx = (int*)ws;    ws += B_ * sizeof(int);
  int*   ntgt   = (int*)ws;    ws += B_ * sizeof(int);
  float* h      = (float*)ws;  ws += (size_t)B_ * D2_ * sizeof(float);
  float* pred   = (float*)ws;  ws += (size_t)B_ * D_  * sizeof(float);

  // zero accumulation targets (colsum + numctx + ntgt are contiguous)
  hipMemsetAsync(colsum, 0, (size_t)B_ * D_ * sizeof(float) + 2 * B_ * sizeof(int), stream);

  ntgt_kernel<<<B_, 256, 0, stream>>>(mask, ntgt);
  pool_fused_kernel<<<dim3(B_, L_ / ROWS_), 256, 0, stream>>>(x, colsum, numctx);

  // GEMM1: (32/16)*(2048/16) = 256 waves -> 32 blocks of 256
  gemm1_gelu_kernel<<<32, 256, 0, stream>>>(colsum, numctx, W1, b1, h);
  layernorm_kernel<<<B_, 256, 0, stream>>>(h, gamma, beta);
  // GEMM2: (32/16)*(1024/16) = 128 waves -> 16 blocks of 256
  gemm2_kernel<<<16, 256, 0, stream>>>(h, W2, b2, pred);

  int T = out_size / (B_ * D_);
  if (T < 1) T = 1;
  int total4 = out_size >> 2;
  scatter_kernel<<<(total4 + 255) / 256, 256, 0, stream>>>(pred, ntgt, (v4f*)d_out, T, total4);
}